// ZeroProbBranch_50337016709814
// MI455X (gfx1250) — compile-verified
//
#include <hip/hip_runtime.h>
#include <math.h>

#define NODES_DEF 100000
#define IN_C 128
#define HID 32
#define OUT_C 32
#define NEG_SLOPE 0.2f

typedef __attribute__((ext_vector_type(2))) float v2f;
typedef __attribute__((ext_vector_type(8))) float v8f;

// ---------------------------------------------------------------------------
// WMMA fp32 GEMM: C[M,NC] = A[M,K] x B[K,NC]  (+ optional bias+sigmoid)
// One wave32 computes one 16x16 output tile with V_WMMA_F32_16X16X4_F32.
// Requires M%16==0, NC%16==0, K%4==0 (true here: M=100000, NC=32, K=128/32).
// ---------------------------------------------------------------------------
template <int K, int NC, int EPI> // EPI: 0 = plain store, 1 = +bias, sigmoid
__global__ void gat_gemm_wmma(const float* __restrict__ A,
                              const float* __restrict__ B,
                              const float* __restrict__ bias,
                              float* __restrict__ C, int M)
{
    const int wavesPerBlock = blockDim.x >> 5;
    const int wave = blockIdx.x * wavesPerBlock + (threadIdx.x >> 5);
    const int nTiles = NC / 16;
    const int totalTiles = (M / 16) * nTiles;
    if (wave >= totalTiles) return;            // wave-uniform: EXEC stays all-1s

    const int mTile = wave / nTiles;
    const int nTile = wave % nTiles;
    const int lane  = threadIdx.x & 31;
    const int half  = lane >> 4;               // 0: lanes 0-15, 1: lanes 16-31
    const int r     = lane & 15;
    const int row   = mTile * 16 + r;          // A row for this half-lane
    const int col   = nTile * 16 + r;          // B/C column for this half-lane

    v8f acc = {};
#pragma unroll
    for (int k = 0; k < K; k += 4) {
        // A 16x4 fragment: v0 = K=k+2h, v1 = K=k+2h+1  (8B contiguous load)
        v2f a = *(const v2f*)(A + (size_t)row * K + k + 2 * half);
        // B 4x16 fragment: v0 = B[k+2h][col], v1 = B[k+2h+1][col]
        v2f b;
        b.x = B[(k + 2 * half + 0) * NC + col];
        b.y = B[(k + 2 * half + 1) * NC + col];
        acc = __builtin_amdgcn_wmma_f32_16x16x4_f32(
            /*neg_a=*/false, a, /*neg_b=*/false, b,
            /*c_mod=*/(short)0, acc, /*reuse_a=*/false, /*reuse_b=*/false);
    }

#pragma unroll
    for (int i = 0; i < 8; ++i) {              // C VGPR i -> row i + 8*half
        const int orow = mTile * 16 + i + 8 * half;
        float v = acc[i];
        if (EPI == 1) {
            v += bias[col];
            v = 1.0f / (1.0f + __expf(-v));    // sigmoid
        }
        C[(size_t)orow * NC + col] = v;
    }
}

// ---------------------------------------------------------------------------
// Order-preserving int encoding of f32 for atomicMax (self-inverse).
// ---------------------------------------------------------------------------
__device__ __forceinline__ int ord_f32(float f)
{
    int i = __float_as_int(f);
    return i >= 0 ? i : (i ^ 0x7fffffff);
}
__device__ __forceinline__ float unord_f32(int i)
{
    return __int_as_float(i >= 0 ? i : (i ^ 0x7fffffff));
}
__device__ __forceinline__ float leaky(float x)
{
    return x > 0.0f ? x : NEG_SLOPE * x;
}

// Per-node: attention scores s_src/s_dst = h . a_src / a_dst; reset m/z/agg.
__global__ void gat_node_prep(const float* __restrict__ h,
                              const float* __restrict__ a_src,
                              const float* __restrict__ a_dst,
                              float* __restrict__ s_src,
                              float* __restrict__ s_dst,
                              int*   __restrict__ m_ord,
                              float* __restrict__ z,
                              float* __restrict__ agg,
                              int n)
{
    const int i = blockIdx.x * blockDim.x + threadIdx.x;
    if (i >= n) return;
    const float* hp = h + (size_t)i * HID;
    float* ap = agg + (size_t)i * HID;
    float ss = 0.0f, sd = 0.0f;
#pragma unroll
    for (int c = 0; c < HID; ++c) {
        const float v = hp[c];
        ss += v * a_src[c];
        sd += v * a_dst[c];
        ap[c] = 0.0f;
    }
    s_src[i] = ss;
    s_dst[i] = sd;
    m_ord[i] = (int)0x80000000;   // below every encoded finite float
    z[i]     = 0.0f;
}

__device__ __forceinline__ void edge_endpoints(const int* __restrict__ ei,
                                               int idx, int n_edges,
                                               int& s, int& d)
{
    if (idx < n_edges) { s = ei[idx]; d = ei[n_edges + idx]; }
    else               { s = d = idx - n_edges; }   // implicit self-loops
}

// Pass 1: segment max of leaky-relu edge score over incoming edges of dst.
__global__ void gat_edge_max(const int* __restrict__ ei,
                             const float* __restrict__ s_src,
                             const float* __restrict__ s_dst,
                             int* __restrict__ m_ord,
                             int n_edges, int e_total)
{
    const int idx = blockIdx.x * blockDim.x + threadIdx.x;
    if (idx >= e_total) return;
    int s, d;
    edge_endpoints(ei, idx, n_edges, s, d);
    const float e = leaky(s_src[s] + s_dst[d]);
    atomicMax(&m_ord[d], ord_f32(e));
}

// Pass 2: z[dst] += exp(e - m[dst])
__global__ void gat_edge_sum(const int* __restrict__ ei,
                             const float* __restrict__ s_src,
                             const float* __restrict__ s_dst,
                             const int* __restrict__ m_ord,
                             float* __restrict__ z,
                             int n_edges, int e_total)
{
    const int idx = blockIdx.x * blockDim.x + threadIdx.x;
    if (idx >= e_total) return;
    int s, d;
    edge_endpoints(ei, idx, n_edges, s, d);
    const float e  = leaky(s_src[s] + s_dst[d]);
    const float ex = __expf(e - unord_f32(m_ord[d]));
    atomicAdd(&z[d], ex);
}

// Pass 3: agg[dst] += alpha * h[src]   (dominant pass: 128B gather + 128B RMW)
__global__ void gat_edge_agg(const int* __restrict__ ei,
                             const float* __restrict__ s_src,
                             const float* __restrict__ s_dst,
                             const int* __restrict__ m_ord,
                             const float* __restrict__ z,
                             const float* __restrict__ h,
                             float* __restrict__ agg,
                             int n_edges, int e_total)
{
    const int idx = blockIdx.x * blockDim.x + threadIdx.x;
    if (idx >= e_total) return;
    int s, d;
    edge_endpoints(ei, idx, n_edges, s, d);
    const float e     = leaky(s_src[s] + s_dst[d]);
    const float alpha = __expf(e - unord_f32(m_ord[d])) / (z[d] + 1e-16f);

    const float4* hp = (const float4*)(h + (size_t)s * HID);
    float* ap = agg + (size_t)d * HID;
#pragma unroll
    for (int c4 = 0; c4 < HID / 4; ++c4) {
        const float4 v = hp[c4];
        atomicAdd(ap + 4 * c4 + 0, alpha * v.x);
        atomicAdd(ap + 4 * c4 + 1, alpha * v.y);
        atomicAdd(ap + 4 * c4 + 2, alpha * v.z);
        atomicAdd(ap + 4 * c4 + 3, alpha * v.w);
    }
}

// g = relu(agg + b)
__global__ void gat_bias_relu(const float* __restrict__ agg,
                              const float* __restrict__ bias,
                              float* __restrict__ g, int total)
{
    const int i = blockIdx.x * blockDim.x + threadIdx.x;
    if (i >= total) return;
    const float v = agg[i] + bias[i & (HID - 1)];
    g[i] = v > 0.0f ? v : 0.0f;
}

// ---------------------------------------------------------------------------
extern "C" void kernel_launch(void* const* d_in, const int* in_sizes, int n_in,
                              void* d_out, int out_size, void* d_ws, size_t ws_size,
                              hipStream_t stream)
{
    const float* x      = (const float*)d_in[0];
    const int*   ei     = (const int*)  d_in[1];
    const float* W1     = (const float*)d_in[2];
    const float* a_src1 = (const float*)d_in[3];
    const float* a_dst1 = (const float*)d_in[4];
    const float* b1     = (const float*)d_in[5];
    const float* W2     = (const float*)d_in[6];
    const float* a_src2 = (const float*)d_in[7];
    const float* a_dst2 = (const float*)d_in[8];
    const float* b2     = (const float*)d_in[9];
    const float* Wc     = (const float*)d_in[10];
    const float* bc     = (const float*)d_in[11];
    float* out = (float*)d_out;

    const int n       = in_sizes[0] / IN_C;      // 100000
    const int n_edges = in_sizes[1] / 2;         // 3200000
    const int e_total = n_edges + n;             // + self loops

    // workspace layout (floats): h | g | agg | s_src | s_dst | z | m_ord
    float* ws    = (float*)d_ws;
    float* h     = ws;
    float* g     = h   + (size_t)n * HID;
    float* agg   = g   + (size_t)n * HID;
    float* s_src = agg + (size_t)n * HID;
    float* s_dst = s_src + n;
    float* z     = s_dst + n;
    int*   m_ord = (int*)(z + n);

    const int TB = 256;
    const int nodeBlocks = (n + TB - 1) / TB;
    const int edgeBlocks = (e_total + TB - 1) / TB;
    const int elemBlocks = (n * HID + TB - 1) / TB;
    const int gemmWaves  = (n / 16) * (HID / 16);          // 12500 tiles
    const int gemmBlocks = (gemmWaves + 7) / 8;            // 8 waves/block

    // ---- Layer 1: GAT(128 -> 32) -----------------------------------------
    gat_gemm_wmma<IN_C, HID, 0><<<gemmBlocks, TB, 0, stream>>>(x, W1, nullptr, h, n);
    gat_node_prep<<<nodeBlocks, TB, 0, stream>>>(h, a_src1, a_dst1, s_src, s_dst,
                                                 m_ord, z, agg, n);
    gat_edge_max<<<edgeBlocks, TB, 0, stream>>>(ei, s_src, s_dst, m_ord, n_edges, e_total);
    gat_edge_sum<<<edgeBlocks, TB, 0, stream>>>(ei, s_src, s_dst, m_ord, z, n_edges, e_total);
    gat_edge_agg<<<edgeBlocks, TB, 0, stream>>>(ei, s_src, s_dst, m_ord, z, h, agg,
                                                n_edges, e_total);
    gat_bias_relu<<<elemBlocks, TB, 0, stream>>>(agg, b1, g, n * HID);

    // ---- Layer 2: GAT(32 -> 32) ------------------------------------------
    gat_gemm_wmma<HID, HID, 0><<<gemmBlocks, TB, 0, stream>>>(g, W2, nullptr, h, n);
    gat_node_prep<<<nodeBlocks, TB, 0, stream>>>(h, a_src2, a_dst2, s_src, s_dst,
                                                 m_ord, z, agg, n);
    gat_edge_max<<<edgeBlocks, TB, 0, stream>>>(ei, s_src, s_dst, m_ord, n_edges, e_total);
    gat_edge_sum<<<edgeBlocks, TB, 0, stream>>>(ei, s_src, s_dst, m_ord, z, n_edges, e_total);
    gat_edge_agg<<<edgeBlocks, TB, 0, stream>>>(ei, s_src, s_dst, m_ord, z, h, agg,
                                                n_edges, e_total);
    gat_bias_relu<<<elemBlocks, TB, 0, stream>>>(agg, b2, g, n * HID);

    // ---- Classifier: sigmoid(g @ Wc + bc) --------------------------------
    gat_gemm_wmma<HID, OUT_C, 1><<<gemmBlocks, TB, 0, stream>>>(g, Wc, bc, out, n);
}